// Max_29094108463828
// MI455X (gfx1250) — compile-verified
//
#include <hip/hip_runtime.h>
#include <hip/hip_bf16.h>
#include <math.h>

#define SM_BETA 100.0f

typedef float v4f __attribute__((ext_vector_type(4)));

// Monotone float -> uint32 key: preserves total order of all non-NaN floats
// (incl. +-0, +-inf), so unsigned max on keys == float max.
__device__ __forceinline__ unsigned fkey(float f) {
  unsigned u = __float_as_uint(f);
  return u ^ (unsigned)(((int)u >> 31) | 0x80000000);
}
__device__ __forceinline__ float funkey(unsigned k) {
  unsigned u = k ^ (unsigned)(((int)(~k) >> 31) | 0x80000000);
  return __uint_as_float(u);
}
#define FKEY_NEG_INF 0x007FFFFFu  // fkey(-inf)

// Pass 0: Mv keys = fkey(-inf), denom = 0
__global__ __launch_bounds__(256) void k_init(unsigned* __restrict__ MvKey,
                                              float* __restrict__ denom,
                                              int n) {
  int i = blockIdx.x * 256 + threadIdx.x;
  if (i < n) {
    MvKey[i] = FKEY_NEG_INF;
    denom[i] = 0.0f;
  }
}

// Pass 1: scatter-max as a single uniform global_atomic_max_u32 per element.
// Mv is 12.8MB -> L2-resident; M stream is non-temporal to not evict it.
__global__ __launch_bounds__(256) void k_scatter_max(
    const float* __restrict__ M, const int* __restrict__ dest,
    unsigned* __restrict__ MvKey, int nquads) {
  int i = blockIdx.x * 256 + threadIdx.x;
  if (i >= nquads) return;
  int e = i >> 4;
  int q = i & 15;
  v4f m = __builtin_nontemporal_load((const v4f*)M + i);
  int dst = dest[e];
  unsigned* p = MvKey + (size_t)dst * 64 + (size_t)q * 4;
  atomicMax(p + 0, fkey(m.x));
  atomicMax(p + 1, fkey(m.y));
  atomicMax(p + 2, fkey(m.z));
  atomicMax(p + 3, fkey(m.w));
}

// Pass 1b: decode keys back to float in place (3.2M elems, negligible).
__global__ __launch_bounds__(256) void k_decode(float* __restrict__ Mv,
                                                int n) {
  int i = blockIdx.x * 256 + threadIdx.x;
  if (i < n) {
    unsigned k = ((const unsigned*)Mv)[i];
    Mv[i] = funkey(k);  // empty segment -> -inf (segment_max identity)
  }
}

// Pass 2: z = exp(BETA*m - BETA*Mv[dest]); store z into W; denom[dest] += z.
// M is streamed via double-buffered async global->LDS loads (ASYNCcnt path):
// each wave prefetches tile k+1 while the exp/atomic work of tile k runs.
// Tile = 16 edges = 1024 floats = 4KB (one b128 per lane per tile).
__global__ __launch_bounds__(256) void k_exp_scatter_sum(
    const float* __restrict__ M, const int* __restrict__ dest,
    const float* __restrict__ Mv, float* __restrict__ W,
    float* __restrict__ denom, int E, int nTiles) {
  __shared__ float lds[2][1024];  // 8KB double buffer
  const int t = threadIdx.x;      // 0..255

  // Low 32 bits of a flat shared-pointer are the LDS byte offset (ISA 10.2:
  // LDS aperture maps addr[31:0] -> LDS offset).
  unsigned ldsOff0 = (unsigned)(size_t)(&lds[0][t * 4]);
  unsigned ldsOff1 = (unsigned)(size_t)(&lds[1][t * 4]);

  const size_t maxBase = (size_t)E * 64 - 4;  // clamp: issue stays wave-uniform
  const int stride = gridDim.x;
  int tile = blockIdx.x;

  // Prologue: async-load the first tile into buffer 0 (non-temporal stream).
  if (tile < nTiles) {
    size_t base = (size_t)tile * 1024 + (size_t)t * 4;
    if (base > maxBase) base = maxBase;
    const float* g = M + base;
    asm volatile("global_load_async_to_lds_b128 %0, %1, off th:TH_LOAD_NT"
                 :
                 : "v"(ldsOff0), "v"(g)
                 : "memory");
  }

  int buf = 0;
  for (; tile < nTiles; tile += stride, buf ^= 1) {
    int nextTile = tile + stride;
    if (nextTile < nTiles) {
      size_t base = (size_t)nextTile * 1024 + (size_t)t * 4;
      if (base > maxBase) base = maxBase;
      const float* g = M + base;
      unsigned off = buf ? ldsOff0 : ldsOff1;
      asm volatile("global_load_async_to_lds_b128 %0, %1, off th:TH_LOAD_NT"
                   :
                   : "v"(off), "v"(g)
                   : "memory");
      // 2 outstanding (current + next); current tile done when <= 1 remains
      // (async loads complete in order within a wave).
      asm volatile("s_wait_asynccnt 0x1" ::: "memory");
    } else {
      asm volatile("s_wait_asynccnt 0x0" ::: "memory");
    }

    int e = tile * 16 + (t >> 4);
    if (e < E) {
      int q = t & 15;
      const float* s = buf ? &lds[1][t * 4] : &lds[0][t * 4];
      v4f m = *(const v4f*)s;  // ds_load_b128
      int dst = dest[e];
      const v4f mv = *(const v4f*)(Mv + (size_t)dst * 64 + (size_t)q * 4);
      // Match reference rounding: logits = 100*m, seg_max = 100*mv.
      v4f z;
      z.x = expf(SM_BETA * m.x - SM_BETA * mv.x);
      z.y = expf(SM_BETA * m.y - SM_BETA * mv.y);
      z.z = expf(SM_BETA * m.z - SM_BETA * mv.z);
      z.w = expf(SM_BETA * m.w - SM_BETA * mv.w);
      __builtin_nontemporal_store(z,
                                  (v4f*)(W + (size_t)e * 64 + (size_t)q * 4));
      float* dn = denom + (size_t)dst * 64 + (size_t)q * 4;
      atomicAdd(dn + 0, z.x);  // global_atomic_add_f32, no-return -> STOREcnt
      atomicAdd(dn + 1, z.y);
      atomicAdd(dn + 2, z.z);
      atomicAdd(dn + 3, z.w);
    }
  }
}

// Pass 3: w = z / denom[dest]. Non-temporal RMW of the 320MB w buffer; denom
// gathers hit L2.
__global__ __launch_bounds__(256) void k_normalize(
    float* __restrict__ W, const int* __restrict__ dest,
    const float* __restrict__ denom, int nquads) {
  int i = blockIdx.x * 256 + threadIdx.x;
  if (i >= nquads) return;
  int e = i >> 4;
  int q = i & 15;
  int dst = dest[e];
  v4f z = __builtin_nontemporal_load((const v4f*)W + i);
  const v4f dn = *(const v4f*)(denom + (size_t)dst * 64 + (size_t)q * 4);
  z.x /= dn.x;
  z.y /= dn.y;
  z.z /= dn.z;
  z.w /= dn.w;
  __builtin_nontemporal_store(z, (v4f*)W + i);
}

extern "C" void kernel_launch(void* const* d_in, const int* in_sizes, int n_in,
                              void* d_out, int out_size, void* d_ws,
                              size_t ws_size, hipStream_t stream) {
  const float* M = (const float*)d_in[0];
  const int* dest = (const int*)d_in[1];
  // d_in[2] (dim_size) lives on device; derive N from shapes instead:
  // out_size = N*d + E*d.
  const int E = in_sizes[1];
  const int d = in_sizes[0] / E;  // 64 for this problem (pass 2 assumes 64)
  const int N = out_size / d - E;

  float* Mv = (float*)d_out;                 // [N, d]
  float* W = (float*)d_out + (size_t)N * d;  // [E, 1, d]
  float* denom = (float*)d_ws;               // [N, d] scratch (12.8 MB)

  const int nSeg = N * d;
  const int nquads = E * (d / 4);
  const int nTiles = (E + 15) / 16;

  k_init<<<(nSeg + 255) / 256, 256, 0, stream>>>((unsigned*)Mv, denom, nSeg);
  k_scatter_max<<<(nquads + 255) / 256, 256, 0, stream>>>(M, dest,
                                                          (unsigned*)Mv,
                                                          nquads);
  k_decode<<<(nSeg + 255) / 256, 256, 0, stream>>>(Mv, nSeg);
  int blocks2 = nTiles < 8192 ? nTiles : 8192;  // >=2 tiles/block -> pipelined
  k_exp_scatter_sum<<<blocks2, 256, 0, stream>>>(M, dest, Mv, W, denom, E,
                                                 nTiles);
  k_normalize<<<(nquads + 255) / 256, 256, 0, stream>>>(W, dest, denom,
                                                        nquads);
}